// EEGGraphModel_1640677507490
// MI455X (gfx1250) — compile-verified
//
#include <hip/hip_runtime.h>
#include <hip/hip_bf16.h>
#include <math.h>

typedef __attribute__((ext_vector_type(2))) float v2f;
typedef __attribute__((ext_vector_type(8))) float v8f;

// CDNA5 fp32 WMMA: D(16x16) = A(16x4) * B(4x16) + C, wave32.
__device__ __forceinline__ v8f wmma_f32(v2f a, v2f b, v8f c) {
  return __builtin_amdgcn_wmma_f32_16x16x4_f32(
      /*neg_a=*/false, a, /*neg_b=*/false, b,
      /*c_mod=*/(short)0, c, /*reuse_a=*/false, /*reuse_b=*/false);
}

// Fast transcendentals for the sequential LSTM critical path:
// hardware v_rcp_f32 instead of the IEEE div_scale/div_fixup sequence.
__device__ __forceinline__ float fast_rcp(float x) {
  return __builtin_amdgcn_rcpf(x);
}
__device__ __forceinline__ float sigmoidf_(float x) {
  return fast_rcp(1.0f + __expf(-x));       // exp(-x)->inf => rcp->0 (ok)
}
__device__ __forceinline__ float tanhf_(float x) {
  // tanh(x) = 1 - 2/(1+exp(2x)); exp overflow -> +1, underflow -> -1.
  return 1.0f - 2.0f * fast_rcp(1.0f + __expf(2.0f * x));
}

// ---------------------------------------------------------------------------
// Kernel 1: fused conv1(8,1,11,s5,p5) -> relu -> conv2(16,8,7,s5,p3) -> relu
//           -> conv3(8,16,5,s5,p2) -> relu, all through LDS.
// Output layout: x_lstm[n][t3][c3]  (256 x 2000 x 8), matches swapaxes(1,2).
// ---------------------------------------------------------------------------
#define TILE3 16
#define N2T   (5 * TILE3)         // 80
#define N1T   (25 * TILE3 + 2)    // 402
#define NINT  (125 * TILE3 + 16)  // 2016

__global__ void conv_fused_kernel(const float* __restrict__ x,     // 256 x 250000
                                  const float* __restrict__ w1,    // 8x1x11
                                  const float* __restrict__ b1,    // 8
                                  const float* __restrict__ w2,    // 16x8x7
                                  const float* __restrict__ b2,    // 16
                                  const float* __restrict__ w3,    // 8x16x5
                                  const float* __restrict__ b3,    // 8
                                  float* __restrict__ out)         // 256x2000x8
{
  __shared__ float s_in[NINT];
  __shared__ float s_y1[8][N1T];
  __shared__ float s_y2[16][N2T];
  __shared__ float s_w1[8 * 11];
  __shared__ float s_w2[16 * 8 * 7];
  __shared__ float s_w3[8 * 16 * 5];

  const int n    = blockIdx.x;       // channel 0..255
  const int t3_0 = blockIdx.y * TILE3;
  const int tid  = threadIdx.x;
  const int NT   = blockDim.x;

  for (int i = tid; i < 8 * 11; i += NT)      s_w1[i] = w1[i];
  for (int i = tid; i < 16 * 8 * 7; i += NT)  s_w2[i] = w2[i];
  for (int i = tid; i < 8 * 16 * 5; i += NT)  s_w3[i] = w3[i];

  // Input window: first needed raw sample is 125*t3_0 - 70.
  const int in0 = 125 * t3_0 - 70;
  const float* xn = x + n * 250000;
  for (int i = tid; i < NINT; i += NT) {
    const int gi = in0 + i;
    s_in[i] = (gi >= 0 && gi < 250000) ? xn[gi] : 0.0f;
  }
  __syncthreads();

  // conv1: t1 window [t1_0, t1_0 + N1T)
  const int t1_0 = 25 * t3_0 - 13;
  for (int i = tid; i < 8 * N1T; i += NT) {
    const int c1 = i / N1T;
    const int it = i - c1 * N1T;
    const int t1 = t1_0 + it;
    float v = 0.0f;
    if (t1 >= 0 && t1 < 50000) {
      float a = b1[c1];
      const int base = 5 * it;  // (5*t1 - 5) - in0 == 5*it
      #pragma unroll
      for (int k = 0; k < 11; ++k) a += s_w1[c1 * 11 + k] * s_in[base + k];
      v = fmaxf(a, 0.0f);
    }
    s_y1[c1][it] = v;
  }
  __syncthreads();

  // conv2: t2 window [t2_0, t2_0 + N2T)
  const int t2_0 = 5 * t3_0 - 2;
  for (int i = tid; i < 16 * N2T; i += NT) {
    const int c2  = i / N2T;
    const int it2 = i - c2 * N2T;
    const int t2  = t2_0 + it2;
    float v = 0.0f;
    if (t2 >= 0 && t2 < 10000) {
      float a = b2[c2];
      #pragma unroll
      for (int c1 = 0; c1 < 8; ++c1) {
        #pragma unroll
        for (int k = 0; k < 7; ++k)
          a += s_w2[(c2 * 8 + c1) * 7 + k] * s_y1[c1][5 * it2 + k];
      }
      v = fmaxf(a, 0.0f);
    }
    s_y2[c2][it2] = v;
  }
  __syncthreads();

  // conv3 + store as [n][t3][c3]
  for (int i = tid; i < TILE3 * 8; i += NT) {
    const int it3 = i >> 3;
    const int c3  = i & 7;
    const int t3  = t3_0 + it3;
    float a = b3[c3];
    #pragma unroll
    for (int c2 = 0; c2 < 16; ++c2) {
      #pragma unroll
      for (int k = 0; k < 5; ++k)
        a += s_w3[(c3 * 16 + c2) * 5 + k] * s_y2[c2][5 * it3 + k];
    }
    out[(n * 2000 + t3) * 8 + c3] = fmaxf(a, 0.0f);
  }
}

// ---------------------------------------------------------------------------
// Kernel 2: LSTM (input 8, hidden 16, T=2000) via fp32 WMMA.
// One workgroup = 16 sequences; 4 waves own gate groups i/f/g/o.
// Per step: gates(16x64) = x_t(16x8)@Wih^T + h(16x16)@Whh^T + b  -> 6 WMMAs/wave.
// Weights live in VGPRs as B fragments for all 2000 steps.
// ---------------------------------------------------------------------------
__global__ void lstm_kernel(const float* __restrict__ x,      // 256 x 2000 x 8
                            const float* __restrict__ w_ih,   // 64 x 8
                            const float* __restrict__ w_hh,   // 64 x 16
                            const float* __restrict__ b_ih,   // 64
                            const float* __restrict__ b_hh,   // 64
                            float* __restrict__ h_out)        // 256 x 16 (tanh'd)
{
  __shared__ float s_h[16][16];        // [seq][hid]
  __shared__ float s_g[4][16][16];     // activated gate tiles [gate][seq][hid]

  const int tid  = threadIdx.x;        // 0..127
  const int wave = tid >> 5;           // gate group: 0=i 1=f 2=g 3=o
  const int lane = tid & 31;
  const int s0   = blockIdx.x * 16;

  const int nn = lane & 15;            // N (gate-within-group) / M (seq) index
  const int kh = (lane >> 4) * 2;      // K sub-offset per lane half

  // Resident B fragments (weights, constant over time).
  const int g = wave * 16 + nn;        // absolute gate row
  v2f bx[2], bh[4];
  #pragma unroll
  for (int s = 0; s < 2; ++s) {
    bx[s].x = w_ih[g * 8 + s * 4 + kh];
    bx[s].y = w_ih[g * 8 + s * 4 + kh + 1];
  }
  #pragma unroll
  for (int s = 0; s < 4; ++s) {
    bh[s].x = w_hh[g * 16 + s * 4 + kh];
    bh[s].y = w_hh[g * 16 + s * 4 + kh + 1];
  }
  const float bias = b_ih[g] + b_hh[g];
  v8f cinit;
  #pragma unroll
  for (int r = 0; r < 8; ++r) cinit[r] = bias;

  v8f c_state = {};                    // cell state (meaningful in wave 0)

  for (int i = tid; i < 256; i += blockDim.x) s_h[i >> 4][i & 15] = 0.0f;
  __syncthreads();

  const int jj = lane & 15;            // N position in D tile
  const int sb = (lane >> 4) * 8;      // seq base in D tile

  for (int t = 0; t < 2000; ++t) {
    // A fragments: x part (K=0..7) from global, h part (K=0..15) from LDS.
    const float* xrow = x + ((s0 + nn) * 2000 + t) * 8;
    v2f ax[2], ah[4];
    #pragma unroll
    for (int s = 0; s < 2; ++s) {
      ax[s].x = xrow[s * 4 + kh];
      ax[s].y = xrow[s * 4 + kh + 1];
    }
    #pragma unroll
    for (int s = 0; s < 4; ++s) {
      ah[s].x = s_h[nn][s * 4 + kh];
      ah[s].y = s_h[nn][s * 4 + kh + 1];
    }

    v8f acc = cinit;
    acc = wmma_f32(ax[0], bx[0], acc);
    acc = wmma_f32(ax[1], bx[1], acc);
    #pragma unroll
    for (int s = 0; s < 4; ++s) acc = wmma_f32(ah[s], bh[s], acc);

    // Activations: waves 0,1,3 sigmoid; wave 2 tanh (fast v_rcp/v_exp path).
    v8f act;
    if (wave == 2) {
      #pragma unroll
      for (int r = 0; r < 8; ++r) act[r] = tanhf_(acc[r]);
    } else {
      #pragma unroll
      for (int r = 0; r < 8; ++r) act[r] = sigmoidf_(acc[r]);
    }
    #pragma unroll
    for (int r = 0; r < 8; ++r) s_g[wave][sb + r][jj] = act[r];
    __syncthreads();

    if (wave == 0) {  // combine: c = f*c + i*g ; h = o*tanh(c)
      #pragma unroll
      for (int r = 0; r < 8; ++r) {
        const float iv = act[r];
        const float fv = s_g[1][sb + r][jj];
        const float gv = s_g[2][sb + r][jj];
        const float ov = s_g[3][sb + r][jj];
        const float c  = fv * c_state[r] + iv * gv;
        c_state[r] = c;
        s_h[sb + r][jj] = ov * tanhf_(c);
      }
    }
    __syncthreads();
  }

  // h_n = tanh(last hidden)
  for (int i = tid; i < 256; i += blockDim.x)
    h_out[(s0 + (i >> 4)) * 16 + (i & 15)] = tanhf_(s_h[i >> 4][i & 15]);
}

// ---------------------------------------------------------------------------
// Kernel 3: feature-mean centering  m = h_n - mean(h_n, axis=1)
// ---------------------------------------------------------------------------
__global__ void center_kernel(const float* __restrict__ h, float* __restrict__ m) {
  const int i = blockIdx.x * blockDim.x + threadIdx.x;
  if (i < 256) {
    float s = 0.0f;
    #pragma unroll
    for (int j = 0; j < 16; ++j) s += h[i * 16 + j];
    s *= (1.0f / 16.0f);
    #pragma unroll
    for (int j = 0; j < 16; ++j) m[i * 16 + j] = h[i * 16 + j] - s;
  }
}

// ---------------------------------------------------------------------------
// Kernel 4: cov = m @ m^T (256x256, K=16) via fp32 WMMA. One wave per tile.
// ---------------------------------------------------------------------------
__global__ void cov_kernel(const float* __restrict__ m, float* __restrict__ cov) {
  const int b    = blockIdx.x;            // 0..255 -> (row tile, col tile)
  const int row0 = (b >> 4) * 16;
  const int col0 = (b & 15) * 16;
  const int lane = threadIdx.x;
  const int nn   = lane & 15;
  const int kh   = (lane >> 4) * 2;

  v8f acc = {};
  #pragma unroll
  for (int kb = 0; kb < 16; kb += 4) {
    v2f a, bb;
    a.x  = m[(row0 + nn) * 16 + kb + kh];
    a.y  = m[(row0 + nn) * 16 + kb + kh + 1];
    bb.x = m[(col0 + nn) * 16 + kb + kh];   // B[k][n] = m[col0+n][k]
    bb.y = m[(col0 + nn) * 16 + kb + kh + 1];
    acc = wmma_f32(a, bb, acc);
  }
  const int jj = lane & 15;
  const int sb = (lane >> 4) * 8;
  #pragma unroll
  for (int r = 0; r < 8; ++r)
    cov[(row0 + sb + r) * 256 + col0 + jj] = acc[r];
}

// ---------------------------------------------------------------------------
// Kernel 5: std from diag(cov); deg_i = #{j : |corr_ij| > 0.6}; dis = rsqrt(deg)
// ---------------------------------------------------------------------------
__global__ void deg_kernel(const float* __restrict__ cov,
                           float* __restrict__ stdv, float* __restrict__ dis) {
  __shared__ float s_std[256];
  const int i = threadIdx.x;
  const float sd = sqrtf(cov[i * 256 + i]);
  s_std[i] = sd;
  __syncthreads();
  float deg = 0.0f;
  for (int j = 0; j < 256; ++j) {
    const float corr = cov[i * 256 + j] / (sd * s_std[j]);
    deg += (fabsf(corr) > 0.6f) ? 1.0f : 0.0f;
  }
  stdv[i] = sd;
  dis[i]  = rsqrtf(deg);                 // deg >= 1 (diagonal always passes)
}

// ---------------------------------------------------------------------------
// Kernel 6: a_norm = adj * dis_i * dis_j
// ---------------------------------------------------------------------------
__global__ void anorm_kernel(const float* __restrict__ cov,
                             const float* __restrict__ stdv,
                             const float* __restrict__ dis,
                             float* __restrict__ an) {
  const int idx = blockIdx.x * blockDim.x + threadIdx.x;   // 65536 total
  const int i = idx >> 8, j = idx & 255;
  const float corr = cov[idx] / (stdv[i] * stdv[j]);
  an[idx] = (fabsf(corr) > 0.6f) ? dis[i] * dis[j] : 0.0f;
}

// ---------------------------------------------------------------------------
// Kernel 7: GCN layer  out = relu(a_norm @ (in @ w) + b)  via fp32 WMMA.
// One wave per 16-row output tile; t = in@w computed into LDS per block.
// ---------------------------------------------------------------------------
__global__ void gcn_kernel(const float* __restrict__ an,   // 256x256
                           const float* __restrict__ hin,  // 256x16
                           const float* __restrict__ w,    // 16x16
                           const float* __restrict__ b,    // 16
                           float* __restrict__ hout) {     // 256x16
  __shared__ float s_t[256][16];
  const int lane = threadIdx.x;
  for (int i = lane; i < 256 * 16; i += 32) {
    const int r = i >> 4, c = i & 15;
    float a = 0.0f;
    #pragma unroll
    for (int k = 0; k < 16; ++k) a += hin[r * 16 + k] * w[k * 16 + c];
    s_t[r][c] = a;
  }
  __syncthreads();

  const int row0 = blockIdx.x * 16;
  const int nn   = lane & 15;
  const int kh   = (lane >> 4) * 2;
  v8f acc;
  const float bias = b[nn];
  #pragma unroll
  for (int r = 0; r < 8; ++r) acc[r] = bias;

  for (int kb = 0; kb < 256; kb += 4) {
    v2f a, bb;
    a.x  = an[(row0 + nn) * 256 + kb + kh];
    a.y  = an[(row0 + nn) * 256 + kb + kh + 1];
    bb.x = s_t[kb + kh][nn];
    bb.y = s_t[kb + kh + 1][nn];
    acc = wmma_f32(a, bb, acc);
  }
  const int jj = lane & 15;
  const int sb = (lane >> 4) * 8;
  #pragma unroll
  for (int r = 0; r < 8; ++r)
    hout[(row0 + sb + r) * 16 + jj] = fmaxf(acc[r], 0.0f);
}

// ---------------------------------------------------------------------------
// Kernel 8: pooled = sum over nodes; out = pooled @ cls_w^T + cls_b
// ---------------------------------------------------------------------------
__global__ void head_kernel(const float* __restrict__ h,
                            const float* __restrict__ cls_w,
                            const float* __restrict__ cls_b,
                            float* __restrict__ out) {
  __shared__ float s_p[16];
  const int lane = threadIdx.x;
  if (lane < 16) {
    float s = 0.0f;
    for (int r = 0; r < 256; ++r) s += h[r * 16 + lane];
    s_p[lane] = s;
  }
  __syncthreads();
  if (lane < 2) {
    float a = cls_b[lane];
    #pragma unroll
    for (int k = 0; k < 16; ++k) a += s_p[k] * cls_w[lane * 16 + k];
    out[lane] = a;
  }
}

// ---------------------------------------------------------------------------
extern "C" void kernel_launch(void* const* d_in, const int* in_sizes, int n_in,
                              void* d_out, int out_size, void* d_ws, size_t ws_size,
                              hipStream_t stream) {
  const float* data    = (const float*)d_in[0];   // 256 x 250000
  const float* conv1_w = (const float*)d_in[1];
  const float* conv1_b = (const float*)d_in[2];
  const float* conv2_w = (const float*)d_in[3];
  const float* conv2_b = (const float*)d_in[4];
  const float* conv3_w = (const float*)d_in[5];
  const float* conv3_b = (const float*)d_in[6];
  const float* lstm_wih = (const float*)d_in[7];  // 64x8
  const float* lstm_whh = (const float*)d_in[8];  // 64x16
  const float* lstm_bih = (const float*)d_in[9];
  const float* lstm_bhh = (const float*)d_in[10];
  const float* gnn1_w  = (const float*)d_in[11];
  const float* gnn1_b  = (const float*)d_in[12];
  const float* gnn2_w  = (const float*)d_in[13];
  const float* gnn2_b  = (const float*)d_in[14];
  const float* cls_w   = (const float*)d_in[15];
  const float* cls_b   = (const float*)d_in[16];
  float* out = (float*)d_out;

  // Workspace layout (floats)
  float* ws = (float*)d_ws;
  float* x_lstm = ws;                       size_t off = 256ull * 2000 * 8;
  float* h_n    = ws + off;                 off += 256 * 16;
  float* m      = ws + off;                 off += 256 * 16;
  float* cov    = ws + off;                 off += 256 * 256;
  float* stdv   = ws + off;                 off += 256;
  float* dis    = ws + off;                 off += 256;
  float* an     = ws + off;                 off += 256 * 256;
  float* h1     = ws + off;                 off += 256 * 16;
  float* h2     = ws + off;                 off += 256 * 16;

  conv_fused_kernel<<<dim3(256, 2000 / TILE3), 128, 0, stream>>>(
      data, conv1_w, conv1_b, conv2_w, conv2_b, conv3_w, conv3_b, x_lstm);

  lstm_kernel<<<16, 128, 0, stream>>>(x_lstm, lstm_wih, lstm_whh,
                                      lstm_bih, lstm_bhh, h_n);

  center_kernel<<<1, 256, 0, stream>>>(h_n, m);
  cov_kernel<<<256, 32, 0, stream>>>(m, cov);
  deg_kernel<<<1, 256, 0, stream>>>(cov, stdv, dis);
  anorm_kernel<<<256, 256, 0, stream>>>(cov, stdv, dis, an);

  gcn_kernel<<<16, 32, 0, stream>>>(an, h_n, gnn1_w, gnn1_b, h1);
  gcn_kernel<<<16, 32, 0, stream>>>(an, h1, gnn2_w, gnn2_b, h2);

  head_kernel<<<1, 32, 0, stream>>>(h2, cls_w, cls_b, out);
}